// LFR_64132451664487
// MI455X (gfx1250) — compile-verified
//
#include <hip/hip_runtime.h>
#include <hip/hip_bf16.h>
#include <stdint.h>

// Problem constants fixed by reference setup_inputs()
constexpr int B_    = 64;
constexpr int T_    = 4096;
constexpr int D_    = 80;
constexpr int MM    = 7;
constexpr int NN    = 6;
constexpr int LEFTP = 3;      // ceil((M-1)//2)
constexpr int NFR   = 683;    // (T_all_max - M)//N + 1 with T_all_max = 4099

constexpr int F_PER_BLK  = 16;                                // output frames per block
constexpr int BLKS_PER_B = (NFR + F_PER_BLK - 1) / F_PER_BLK; // 43
constexpr int SM_FRAMES  = NN * F_PER_BLK + 1;                // 97 staged input frames
constexpr int SM_STRIDE  = 84;                                // floats; 336B, 16B aligned
constexpr int D4         = D_ / 4;                            // 20 float4 per frame

typedef int v4i __attribute__((ext_vector_type(4)));

// ---- CDNA5 async global->LDS 16-byte copy -----------------------------------
__device__ __forceinline__ void async_copy16(const float* g, float* l) {
#if defined(__HIP_DEVICE_COMPILE__)
#if __has_builtin(__builtin_amdgcn_global_load_async_to_lds_b128)
  __builtin_amdgcn_global_load_async_to_lds_b128(
      (__attribute__((address_space(1))) v4i*)(uintptr_t)g,
      (__attribute__((address_space(3))) v4i*)(uintptr_t)l,
      0, 0);
#else
  asm volatile("global_load_async_to_lds_b128 %0, %1, off"
               :
               : "v"((uint32_t)(uintptr_t)l), "v"(g)
               : "memory");
#endif
#else
  (void)g; (void)l;
#endif
}

__device__ __forceinline__ void wait_async0() {
#if defined(__HIP_DEVICE_COMPILE__)
#if __has_builtin(__builtin_amdgcn_s_wait_asynccnt)
  __builtin_amdgcn_s_wait_asynccnt(0);
#else
  asm volatile("s_wait_asynccnt 0" ::: "memory");
#endif
#endif
}
// -----------------------------------------------------------------------------

__global__ __launch_bounds__(256) void lfr_kernel(const float* __restrict__ x,
                                                  const int* __restrict__ lens,
                                                  float* __restrict__ out,
                                                  float* __restrict__ new_len) {
  __shared__ __align__(16) float smem[SM_FRAMES * SM_STRIDE];

  const int blk = blockIdx.x;
  const int b   = blk / BLKS_PER_B;
  const int fb  = blk - b * BLKS_PER_B;
  const int f0  = fb * F_PER_BLK;
  const int F   = (NFR - f0 < F_PER_BLK) ? (NFR - f0) : F_PER_BLK;
  const int tid = threadIdx.x;
  const int len = lens[b];

  // new_len[b] = (LEFT + len + right_pad) // N  (exact integer reproduction)
  if (fb == 0 && tid == 0) {
    int q     = (len + NN - 1) / NN;            // ceil(len/N)
    int delta = len + LEFTP + NN - NN * q;      // len + 9 - 6q
    int rp    = (MM - delta > 0) ? (MM - delta) : 0;
    new_len[b] = (float)((LEFTP + len + rp) / NN);
  }

  // ---- Stage (6F+1) input frames into LDS via async copies ----
  const int nfr_lds = NN * F + 1;
  const int nload   = nfr_lds * D4;             // float4 granules
  const float* xb   = x + (size_t)b * T_ * D_;
  const int lim     = LEFTP + len;

  for (int l = tid; l < nload; l += 256) {
    int j  = l / D4;
    int d4 = l - j * D4;
    int i  = f0 * NN + j;                       // padded-timeline index
    int s  = i - LEFTP;
    if (s < 0) s = 0;                           // head: repeat x[0]
    if (i >= lim) s = T_ - 1;                   // beyond len: x[T-1] (ref semantics)
    async_copy16(xb + (size_t)s * D_ + d4 * 4,
                 &smem[j * SM_STRIDE + d4 * 4]);
  }
  wait_async0();
  __syncthreads();

  // ---- Fan out 7-frame windows: out[b, f, m*80+d] = smem[(f-f0)*6+m][d] ----
  const int nstore = F * MM * D4;
  for (int l = tid; l < nstore; l += 256) {
    int fl  = l / (MM * D4);
    int rem = l - fl * (MM * D4);
    int m   = rem / D4;
    int d4  = rem - m * D4;
    const float4 v = *(const float4*)&smem[(fl * NN + m) * SM_STRIDE + d4 * 4];
    size_t o = ((size_t)((b * NFR + f0 + fl) * MM + m)) * D_ + (size_t)d4 * 4;
    *(float4*)&out[o] = v;
  }
}

extern "C" void kernel_launch(void* const* d_in, const int* in_sizes, int n_in,
                              void* d_out, int out_size, void* d_ws, size_t ws_size,
                              hipStream_t stream) {
  const float* x    = (const float*)d_in[0];
  const int*   lens = (const int*)d_in[1];
  float*       out  = (float*)d_out;
  float*       nlen = out + (size_t)B_ * NFR * MM * D_;   // tuple part 2

  dim3 grid(B_ * BLKS_PER_B), block(256);
  lfr_kernel<<<grid, block, 0, stream>>>(x, lens, out, nlen);
}